// TaylorExp_4028679324114
// MI455X (gfx1250) — compile-verified
//
#include <hip/hip_runtime.h>

typedef __attribute__((ext_vector_type(2))) float v2f;
typedef __attribute__((ext_vector_type(8))) float v8f;

namespace {
constexpr int   kD    = 16;
constexpr int   kOutD = 1 + kD + kD * kD;        // 273 floats per token
constexpr float kSx   = 0.5f;                    // 1 / d^0.25  (d=16)
constexpr float kSxx  = 0.17677669529663687f;    // 1 / (sqrt(2) * sqrt(d)) = 1/(4*sqrt(2))
}

// One wave (32 lanes) handles one token:
//   - lanes 0..15 load x[0..15] (coalesced 64B)
//   - one V_WMMA_F32_16X16X4_F32 computes the 16x16 outer product x * x^T
//     (x placed in K=0 column of A and K=0 row of B; K=1..3 zeroed)
//   - D layout: lane = column N (0..15, repeated), VGPR v = row M (lanes 16..31
//     hold rows 8..15), which maps straight onto the row-major d*d flattening.
// All global traffic uses non-temporal hints: the 286MB output is write-once.
__global__ __launch_bounds__(256) void taylor_exp_wmma(const float* __restrict__ x,
                                                       float* __restrict__ out,
                                                       int ntok) {
    const int lane  = threadIdx.x & 31;
    const int wave  = threadIdx.x >> 5;
    const int token = blockIdx.x * 8 + wave;     // 8 waves per 256-thread block
    if (token >= ntok) return;                   // wave-uniform: EXEC stays all-1s for WMMA

    const float* xp = x + (size_t)token * kD;
    float xv = 0.0f;
    if (lane < kD) xv = __builtin_nontemporal_load(xp + lane);

    // A (16x4 f32): lanes 0-15 VGPR0 = K=0 column = x[M]; VGPR1 (K=1) = 0;
    // lanes 16-31 (K=2,3) = 0.  B (4x16 f32): K=0 row = x[N] in VGPR0 lanes 0-15.
    v2f a;
    a.x = (lane < 16) ? xv : 0.0f;
    a.y = 0.0f;
    v8f c = {};
    v8f d = __builtin_amdgcn_wmma_f32_16x16x4_f32(
        /*neg_a=*/false, a, /*neg_b=*/false, a,
        /*c_mod=*/(short)0, c, /*reuse_a=*/false, /*reuse_b=*/false);

    float* op = out + (size_t)token * kOutD;

    // Header: out[0] = 1, out[1+i] = x[i] * 0.5
    if (lane == 0) __builtin_nontemporal_store(1.0f, op);
    if (lane < kD) __builtin_nontemporal_store(xv * kSx, op + 1 + lane);

    // Second-order block: out[17 + M*16 + N] = x[M]*x[N] * kSxx
    const int mBase = (lane >> 4) << 3;   // rows 0..7 (lanes 0-15) or 8..15 (lanes 16-31)
    const int col   = lane & 15;          // column N
    float* q = op + 1 + kD + col;
#pragma unroll
    for (int v = 0; v < 8; ++v) {
        __builtin_nontemporal_store(d[v] * kSxx, q + (size_t)(mBase + v) * kD);
    }
}

extern "C" void kernel_launch(void* const* d_in, const int* in_sizes, int n_in,
                              void* d_out, int out_size, void* d_ws, size_t ws_size,
                              hipStream_t stream) {
    const float* x = (const float*)d_in[0];
    float* out = (float*)d_out;
    const int ntok   = in_sizes[0] / kD;     // b*h*s = 262144 tokens
    const int blocks = (ntok + 7) / 8;       // one token per wave, 8 waves per block
    taylor_exp_wmma<<<blocks, 256, 0, stream>>>(x, out, ntok);
}